// ClassOverlapMaskFilter_53523882442766
// MI455X (gfx1250) — compile-verified
//
#include <hip/hip_runtime.h>

// Sizes fixed by the reference.
#define N_BOXES 1024
#define Hd 512
#define Wd 512
#define IOU_THR 0.8f

typedef __attribute__((ext_vector_type(2))) float f32x2;
typedef __attribute__((ext_vector_type(4))) float f32x4;
typedef __attribute__((ext_vector_type(8))) float f32x8;

// ---------------------------------------------------------------------------
// Pass A: masks -> boxes. One block (8 waves) per mask. Each wave owns two
// 32-column strip-pairs; per 4-row step it loads two 4x16 f32 tiles laid out
// exactly as the WMMA 16x4 f32 A-matrix (lane<16: K=0/1, lane>=16: K=2/3):
//   D += A(mask) x B(ones)   ->  D[m][*] = exact column count (any_x)
// Row presence (any_y): masks are exactly 0/1, so a 2-bit nib per step is
// (unsigned)(ax + 2*ay); nibs pack into a 32-row word, OR-reduced across the
// wave every 8 steps (no per-iteration SALU/exec-mask traffic).
// ---------------------------------------------------------------------------
__global__ void __launch_bounds__(256)
boxes_kernel(const float* __restrict__ masks, float* __restrict__ boxesOut) {
    __shared__ float colSum[Wd];
    __shared__ unsigned rowBits[Hd / 32];
    __shared__ int ibx1, ibx2, iby1, iby2;

    const int tid = threadIdx.x;
    const int n = blockIdx.x;
    if (tid < Hd / 32) rowBits[tid] = 0u;
    if (tid == 0) { ibx1 = Wd; ibx2 = -1; iby1 = Hd; iby2 = -1; }
    __syncthreads();

    const int wave = tid >> 5;
    const int lane = tid & 31;
    const int rOff = (lane >> 4) << 1;           // 0 for lanes 0-15, 2 for 16-31
    const float* m = masks + (size_t)n * (Hd * Wd);
    const f32x2 ones = {1.0f, 1.0f};

    for (int sp = 0; sp < 2; ++sp) {
        const int w0 = (wave * 2 + sp) * 32;     // 8 waves x 2 pairs x 32 cols = 512
        const int c = w0 + (lane & 15);
        f32x8 acc0 = {0.f, 0.f, 0.f, 0.f, 0.f, 0.f, 0.f, 0.f};
        f32x8 acc1 = {0.f, 0.f, 0.f, 0.f, 0.f, 0.f, 0.f, 0.f};

        for (int hb = 0; hb < Hd; hb += 32) {
            unsigned rw = 0u;
#pragma unroll
            for (int t = 0; t < 8; ++t) {
                const int h = hb + t * 4;
                const size_t ro0 = (size_t)(h + rOff) * Wd + c;
                const size_t ro1 = (size_t)(h + rOff + 1) * Wd + c;
                const float ax0 = m[ro0];            // rows h   / h+2, cols w0..+15
                const float ay0 = m[ro1];            // rows h+1 / h+3
                const float ax1 = m[ro0 + 16];       // cols w0+16..+31
                const float ay1 = m[ro1 + 16];
                f32x2 A0 = {ax0, ay0};
                f32x2 A1 = {ax1, ay1};
                acc0 = __builtin_amdgcn_wmma_f32_16x16x4_f32(
                    false, A0, false, ones, (short)0, acc0, false, false);
                acc1 = __builtin_amdgcn_wmma_f32_16x16x4_f32(
                    false, A1, false, ones, (short)0, acc1, false, false);
                // 0/1 masks: nib bit0 = row h+rOff present, bit1 = row h+rOff+1
                const float s = fmaf(fmaxf(ay0, ay1), 2.0f, fmaxf(ax0, ax1));
                rw |= ((unsigned)s) << (t * 4 + rOff);
            }
            // OR-reduce the 32-row word across the wave (both halves merge).
            rw |= __shfl_xor(rw, 1);
            rw |= __shfl_xor(rw, 2);
            rw |= __shfl_xor(rw, 4);
            rw |= __shfl_xor(rw, 8);
            rw |= __shfl_xor(rw, 16);
            if (lane == 0 && rw) atomicOr(&rowBits[hb >> 5], rw);
        }
        // D layout: VGPR r -> (M=r, lanes 0-15) and (M=8+r, lanes 16-31)
        if ((lane & 15) == 0) {
            const int base = w0 + ((lane >> 4) << 3);   // lane0 -> w0, lane16 -> w0+8
#pragma unroll
            for (int r = 0; r < 8; ++r) {
                colSum[base + r]      = acc0[r];
                colSum[base + 16 + r] = acc1[r];
            }
        }
    }
    __syncthreads();

#pragma unroll
    for (int k = 0; k < 2; ++k) {
        const int w = tid + k * 256;
        if (colSum[w] > 0.5f) { atomicMin(&ibx1, w); atomicMax(&ibx2, w); }
        if ((rowBits[w >> 5] >> (w & 31)) & 1u) { atomicMin(&iby1, w); atomicMax(&iby2, w); }
    }
    __syncthreads();
    if (tid == 0) {
        // Reference defaults for empty masks: x1=W, x2=-1, y1=H, y2=-1.
        boxesOut[n * 4 + 0] = (float)ibx1;
        boxesOut[n * 4 + 1] = (float)iby1;
        boxesOut[n * 4 + 2] = (float)ibx2;
        boxesOut[n * 4 + 3] = (float)iby2;
    }
}

// ---------------------------------------------------------------------------
// Pass B1: per-block redundant prep (max_coord, class offsets, stable ranks ==
// argsort(-scores)), then each block fills 64 rows of the 1024x1024 suppression
// bit-matrix in d_ws:   bit(j) of row i  =  (iou(i,j) > thr) && (j > i).
// ---------------------------------------------------------------------------
__global__ void __launch_bounds__(256)
nms_rows_kernel(const float* __restrict__ boxesIn, const float* __restrict__ scores,
                const int* __restrict__ labels, unsigned* __restrict__ matrix,
                int* __restrict__ sortedOrig) {
    __shared__ __align__(16) float sc[N_BOXES];
    __shared__ float4 sb[N_BOXES];
    __shared__ float sa[N_BOXES];
    __shared__ unsigned maxKey;

    const int tid = threadIdx.x;
    if (tid == 0) maxKey = 0u;
#pragma unroll
    for (int k = 0; k < 4; ++k) sc[tid + k * 256] = scores[tid + k * 256];
    __syncthreads();

    float4 myBox[4];
    unsigned localMax = 0u;
#pragma unroll
    for (int k = 0; k < 4; ++k) {
        const int i = tid + k * 256;
        float4 b;
        b.x = boxesIn[i * 4 + 0]; b.y = boxesIn[i * 4 + 1];
        b.z = boxesIn[i * 4 + 2]; b.w = boxesIn[i * 4 + 3];
        myBox[k] = b;
        // order-preserving uint key (all coords >= -1, so +2 keeps them > 0)
        localMax = max(localMax, __float_as_uint(b.x + 2.0f));
        localMax = max(localMax, __float_as_uint(b.y + 2.0f));
        localMax = max(localMax, __float_as_uint(b.z + 2.0f));
        localMax = max(localMax, __float_as_uint(b.w + 2.0f));
    }
    atomicMax(&maxKey, localMax);
    __syncthreads();
    const float maxc = __uint_as_float(maxKey) - 2.0f;

#pragma unroll
    for (int k = 0; k < 4; ++k) {
        const int i = tid + k * 256;
        const float off = (float)labels[i] * (maxc + 1.0f);
        float4 b = myBox[k];
        b.x += off; b.y += off; b.z += off; b.w += off;
        const float si = sc[i];
        int rank = 0;
        for (int j = 0; j < N_BOXES; j += 4) {
            const float4 s4 = ((const float4*)sc)[j >> 2];
            rank += (s4.x > si || (s4.x == si && (j + 0) < i));
            rank += (s4.y > si || (s4.y == si && (j + 1) < i));
            rank += (s4.z > si || (s4.z == si && (j + 2) < i));
            rank += (s4.w > si || (s4.w == si && (j + 3) < i));
        }
        sb[rank] = b;
        sa[rank] = fmaxf(b.z - b.x, 0.f) * fmaxf(b.w - b.y, 0.f);
        if (blockIdx.x == 0) sortedOrig[rank] = i;
    }
    __syncthreads();

    // 64 rows per block, 4 threads per row, 8 words per thread.
    const int r = blockIdx.x * 64 + (tid >> 2);
    const float4 br = sb[r];
    const float ar = sa[r];
    const int wStart = (tid & 3) * 8;
    for (int wj = wStart; wj < wStart + 8; ++wj) {
        unsigned mword = 0u;
        for (int b = 0; b < 32; ++b) {
            const int j = wj * 32 + b;
            const float4 bj = sb[j];
            const float lx = fmaxf(br.x, bj.x), ly = fmaxf(br.y, bj.y);
            const float rx = fminf(br.z, bj.z), ry = fminf(br.w, bj.w);
            const float inter = fmaxf(rx - lx, 0.f) * fmaxf(ry - ly, 0.f);
            const float uni = ar + sa[j] - inter;
            const float iou = inter / fmaxf(uni, 1e-9f);
            if (iou > IOU_THR && j > r) mword |= (1u << b);
        }
        matrix[r * 32 + wj] = mword;
    }
}

// ---------------------------------------------------------------------------
// Pass B2: single wave. 1024-bit keep mask lives in one VGPR across 32 lanes
// (lane L owns sorted indices [32L,32L+32)). Serial scan with double-buffered
// row prefetch (L2-hot). Then scatter keep / points outputs back to original
// order.
// ---------------------------------------------------------------------------
__global__ void __launch_bounds__(32)
nms_scan_kernel(const unsigned* __restrict__ matrix, const int* __restrict__ sortedOrig,
                const float* __restrict__ points, float* __restrict__ keepFlag,
                float* __restrict__ outPoints, float* __restrict__ outKeep) {
    const int lane = threadIdx.x;
    unsigned keep = 0xFFFFFFFFu;
    unsigned bufA[16], bufB[16];
#pragma unroll
    for (int k = 0; k < 16; ++k) bufA[k] = matrix[k * 32 + lane];

    for (int base = 0; base < N_BOXES; base += 16) {
        if (base + 16 < N_BOXES) {
#pragma unroll
            for (int k = 0; k < 16; ++k) bufB[k] = matrix[(base + 16 + k) * 32 + lane];
        }
#pragma unroll
        for (int k = 0; k < 16; ++k) {
            const int i = base + k;
            const unsigned kw = __shfl(keep, i >> 5);
            const unsigned bit = (kw >> (i & 31)) & 1u;
            keep &= ~(bufA[k] & (0u - bit));   // apply row i only if keep[i]
        }
#pragma unroll
        for (int k = 0; k < 16; ++k) bufA[k] = bufB[k];
    }

    for (int b = 0; b < 32; ++b) {
        const int j = b * 32 + lane;           // sorted index, owner lane = b
        const unsigned kw = __shfl(keep, b);
        const float kf = ((kw >> lane) & 1u) ? 1.0f : 0.0f;
        const int orig = sortedOrig[j];
        keepFlag[orig] = kf;
        outKeep[orig] = kf;
        outPoints[orig * 3 + 0] = points[orig * 3 + 0] * kf;
        outPoints[orig * 3 + 1] = points[orig * 3 + 1] * kf;
        outPoints[orig * 3 + 2] = points[orig * 3 + 2] * kf;
    }
}

// ---------------------------------------------------------------------------
// Pass C: masks_kept = masks * keep. Per-block-uniform keep flag; kept masks
// are streamed with non-temporal B128 loads/stores (2 GiB stream must not
// thrash L2); suppressed masks are zero-filled without reading.
// ---------------------------------------------------------------------------
__global__ void __launch_bounds__(256)
filter_kernel(const float* __restrict__ masks, const float* __restrict__ keepFlag,
              float* __restrict__ outMasks) {
    const int n = blockIdx.x >> 2;
    const int chunk = blockIdx.x & 3;
    const size_t base4 = (size_t)n * (Hd * Wd / 4) + (size_t)chunk * (Hd * Wd / 16);
    const f32x4* src = (const f32x4*)masks + base4;
    f32x4* dst = (f32x4*)outMasks + base4;
    const float kf = keepFlag[n];
    const int tid = threadIdx.x;
    if (kf > 0.5f) {
        for (int i = tid; i < Hd * Wd / 16; i += 256) {
            const f32x4 v = __builtin_nontemporal_load(&src[i]);
            __builtin_nontemporal_store(v, &dst[i]);
        }
    } else {
        const f32x4 z = {0.f, 0.f, 0.f, 0.f};
        for (int i = tid; i < Hd * Wd / 16; i += 256) {
            __builtin_nontemporal_store(z, &dst[i]);
        }
    }
}

// ---------------------------------------------------------------------------
extern "C" void kernel_launch(void* const* d_in, const int* in_sizes, int n_in,
                              void* d_out, int out_size, void* d_ws, size_t ws_size,
                              hipStream_t stream) {
    const float* masks  = (const float*)d_in[0];
    const float* scores = (const float*)d_in[1];
    const float* points = (const float*)d_in[2];
    const int*   labels = (const int*)d_in[3];

    float* out = (float*)d_out;
    float* outMasks  = out;                                    // [1024*512*512]
    float* outPoints = out + (size_t)N_BOXES * Hd * Wd;        // [1024*3]
    float* outKeep   = outPoints + N_BOXES * 3;                // [1024]

    char* ws = (char*)d_ws;
    float*    boxes      = (float*)ws;                         // 16 KB
    unsigned* matrix     = (unsigned*)(ws + 16384);            // 128 KB
    int*      sortedOrig = (int*)(ws + 16384 + 131072);        // 4 KB
    float*    keepFlag   = (float*)(ws + 16384 + 131072 + 4096); // 4 KB

    boxes_kernel<<<N_BOXES, 256, 0, stream>>>(masks, boxes);
    nms_rows_kernel<<<16, 256, 0, stream>>>(boxes, scores, labels, matrix, sortedOrig);
    nms_scan_kernel<<<1, 32, 0, stream>>>(matrix, sortedOrig, points, keepFlag,
                                          outPoints, outKeep);
    filter_kernel<<<N_BOXES * 4, 256, 0, stream>>>(masks, keepFlag, outMasks);
}